// KANLayerOld_56925496541762
// MI455X (gfx1250) — compile-verified
//
#include <hip/hip_runtime.h>
#include <math.h>

#define IN_F        64
#define WID         64
#define GRID_PTS    64
#define ROWS_PER_WG 64
#define THREADS     256
#define NWAVES      8
#define RPW         8          // rows per wave
#define LN_EPS      1e-5f

typedef float v2f __attribute__((ext_vector_type(2)));
typedef float v8f __attribute__((ext_vector_type(8)));

// ---- dynamic LDS partition (bytes) ----
// [     0,  65536) : s_w4    float4[64*64]      basis weights
// [ 65536,  98304) : s_stage float[2][64][64]   coeff double buffer / cmat reuse
// [ 98304, 114944) : s_inp   float[64][65]      input tile (padded)
// [114944, 131584) : s_skw   float[64][65]      skip_W tile (padded)
// [131584, 139776) : s_idx   ushort[64*64]      span indices
// [139776, 141824) : params  8 * 64 floats
#define SMEM_BYTES 141824

__device__ __forceinline__ float knotv(int j) {
    // clamped knot vector: 4 zeros, interior k/61 (k=1..60), 4 ones
    float v = (float)(j - 3) * (1.0f / 61.0f);
    return fminf(1.0f, fmaxf(0.0f, v));
}

__global__ void __launch_bounds__(THREADS)
kan_fused_kernel(const float* __restrict__ inputs,
                 const float* __restrict__ in_shift,
                 const float* __restrict__ in_log_scale,
                 const float* __restrict__ coeffs,   // [F][G][W]
                 const float* __restrict__ skip_W,   // [W][F]
                 const float* __restrict__ skip_b,
                 const float* __restrict__ bias,
                 const float* __restrict__ ln_gamma,
                 const float* __restrict__ ln_beta,
                 float* __restrict__ out,
                 int batch)
{
    extern __shared__ __align__(16) char smem_raw[];
    float4*         s_w4    = (float4*)(smem_raw);
    float*          s_stage = (float*)(smem_raw + 65536);
    float*          s_inp   = (float*)(smem_raw + 98304);
    float*          s_skw   = (float*)(smem_raw + 114944);
    unsigned short* s_idx   = (unsigned short*)(smem_raw + 131584);
    float*          s_par   = (float*)(smem_raw + 139776);
    float* s_shift = s_par;        float* s_scale = s_par + 64;
    float* s_skb   = s_par + 128;  float* s_bias  = s_par + 192;
    float* s_gam   = s_par + 256;  float* s_bet   = s_par + 320;
    float* s_mean  = s_par + 384;  float* s_rstd  = s_par + 448;

    const int tid  = threadIdx.x;
    const int wave = tid >> 5;
    const int lane = tid & 31;
    const int row0 = blockIdx.x * ROWS_PER_WG;

    // ---------- Phase A: per-feature params + skip_W staging ----------
    if (tid < 64) {
        float z  = in_log_scale[tid];
        float sp = fmaxf(z, 0.0f) + log1pf(expf(-fabsf(z)));   // stable softplus
        s_scale[tid] = sp + 0.001f;
        s_shift[tid] = in_shift[tid];
        s_skb[tid]   = skip_b[tid];
        s_bias[tid]  = bias[tid];
        s_gam[tid]   = ln_gamma[tid];
        s_bet[tid]   = ln_beta[tid];
    }
    #pragma unroll
    for (int e = 0; e < 16; ++e) {
        int idx = e * THREADS + tid;          // 0..4095
        int n = idx >> 6, k = idx & 63;
        s_skw[n * 65 + k] = skip_W[idx];      // padded stride 65
    }
    __syncthreads();

    // ---------- Phase B: inputs + sparse cubic B-spline basis ----------
    #pragma unroll
    for (int e = 0; e < 16; ++e) {
        int idx = e * THREADS + tid;
        int b = idx >> 6, f = idx & 63;
        int gb = row0 + b;
        float x = (gb < batch) ? inputs[(size_t)gb * IN_F + f] : 0.0f;
        s_inp[b * 65 + f] = x;
        float t  = (x - s_shift[f]) * s_scale[f];
        float sg = 1.0f / (1.0f + expf(-t));          // in (0,1)
        int cell = (int)(sg * 61.0f);
        cell = cell < 0 ? 0 : (cell > 60 ? 60 : cell);
        int i = cell + 3;                             // knot span: knots[i] <= sg < knots[i+1]
        // Cox-de Boor basisFuns (degree 3): 4 non-zero basis values N[0..3]
        float Nv[4] = {1.0f, 0.0f, 0.0f, 0.0f};
        float lft[4], rgt[4];
        #pragma unroll
        for (int d = 1; d <= 3; ++d) {
            lft[d] = sg - knotv(i + 1 - d);
            rgt[d] = knotv(i + d) - sg;
            float saved = 0.0f;
            #pragma unroll
            for (int r = 0; r < d; ++r) {
                float tmp = Nv[r] / (rgt[r + 1] + lft[d - r]);
                Nv[r] = saved + rgt[r + 1] * tmp;
                saved = lft[d - r] * tmp;
            }
            Nv[d] = saved;
        }
        s_w4[idx]  = make_float4(Nv[0], Nv[1], Nv[2], Nv[3]);
        s_idx[idx] = (unsigned short)(i - 3);         // first non-zero basis index
    }

    // ---------- Phase C: sparse spline contraction, double-buffered coeffs ----------
    {
        const float4* g0 = (const float4*)(coeffs);   // feature 0 slab
        float4* st = (float4*)(s_stage);
        #pragma unroll
        for (int q = 0; q < 4; ++q) st[q * THREADS + tid] = g0[q * THREADS + tid];
    }
    __syncthreads();

    float2 acc[RPW];
    #pragma unroll
    for (int r = 0; r < RPW; ++r) { acc[r].x = 0.0f; acc[r].y = 0.0f; }
    const int wbase = 2 * lane;            // lane's width slice
    const int rowb  = wave * RPW;

    for (int f = 0; f < IN_F; ++f) {
        const int cur = f & 1;
        float4 ng[4];
        if (f + 1 < IN_F) {                // prefetch next slab into regs
            const float4* gn = (const float4*)(coeffs + (size_t)(f + 1) * (GRID_PTS * WID));
            #pragma unroll
            for (int q = 0; q < 4; ++q) ng[q] = gn[q * THREADS + tid];
        }
        const float* st = s_stage + cur * (GRID_PTS * WID);
        #pragma unroll
        for (int r = 0; r < RPW; ++r) {
            int row = rowb + r;
            float4 wv = s_w4[row * 64 + f];               // wave-uniform broadcast
            int    i0 = (int)s_idx[row * 64 + f];
            const float* cp = st + i0 * WID + wbase;      // conflict-free b64 reads
            float2 q0 = *(const float2*)(cp);
            float2 q1 = *(const float2*)(cp + 64);
            float2 q2 = *(const float2*)(cp + 128);
            float2 q3 = *(const float2*)(cp + 192);
            acc[r].x = fmaf(wv.x, q0.x, acc[r].x);  acc[r].y = fmaf(wv.x, q0.y, acc[r].y);
            acc[r].x = fmaf(wv.y, q1.x, acc[r].x);  acc[r].y = fmaf(wv.y, q1.y, acc[r].y);
            acc[r].x = fmaf(wv.z, q2.x, acc[r].x);  acc[r].y = fmaf(wv.z, q2.y, acc[r].y);
            acc[r].x = fmaf(wv.w, q3.x, acc[r].x);  acc[r].y = fmaf(wv.w, q3.y, acc[r].y);
        }
        __syncthreads();
        if (f + 1 < IN_F) {
            float4* stn = (float4*)(s_stage + (1 - cur) * (GRID_PTS * WID));
            #pragma unroll
            for (int q = 0; q < 4; ++q) stn[q * THREADS + tid] = ng[q];
        }
        __syncthreads();
    }

    // ---------- Phase D1: spline results -> cmat (reuse stage, stride 65) ----------
    float* cmat = s_stage;                 // 64*65 floats, stage buffers are dead now
    #pragma unroll
    for (int r = 0; r < RPW; ++r) {
        int row = rowb + r;
        cmat[row * 65 + wbase]     = acc[r].x;
        cmat[row * 65 + wbase + 1] = acc[r].y;
    }
    __syncthreads();

    // ---------- Phase D2: skip GEMM via V_WMMA_F32_16X16X4_F32 ----------
    // A = inputs tile (16x64), B = skip_W^T (64x16): B[k][n] = skip_W[n][k]
    // ISA layouts: A VGPR0=K even / VGPR1=K odd, half-wave +2K; C/D VGPR v <-> M=v+8*half
    const int m    = lane & 15;
    const int half = lane >> 4;
    #pragma unroll
    for (int t = 0; t < 2; ++t) {
        int tile = wave * 2 + t;                      // 16 tiles over 8 waves
        int rt = tile >> 2, nt = tile & 3;
        const float* arow = s_inp + (rt * 16 + m) * 65;
        const float* brow = s_skw + (nt * 16 + m) * 65;   // m plays the N role for B
        v8f c = {0.f, 0.f, 0.f, 0.f, 0.f, 0.f, 0.f, 0.f};
        #pragma unroll
        for (int k0 = 0; k0 < IN_F; k0 += 4) {
            int kb = k0 + half * 2;
            v2f a; a.x = arow[kb]; a.y = arow[kb + 1];
            v2f b; b.x = brow[kb]; b.y = brow[kb + 1];
            c = __builtin_amdgcn_wmma_f32_16x16x4_f32(false, a, false, b,
                                                      (short)0, c, false, false);
        }
        int col = nt * 16 + m;
        float sb = s_skb[col] + s_bias[col];
        #pragma unroll
        for (int v = 0; v < 8; ++v) {
            int row = rt * 16 + v + half * 8;
            int adr = row * 65 + col;
            cmat[adr] = cmat[adr] + c[v] + sb;
        }
    }
    __syncthreads();

    // ---------- Phase D3: LayerNorm + exact GELU ----------
    if (tid < 64) {
        const float* crow = cmat + tid * 65;          // stride 65 -> conflict-free
        float mval = 0.0f;
        #pragma unroll
        for (int k = 0; k < 64; ++k) mval += crow[k];
        mval *= (1.0f / 64.0f);
        float vv = 0.0f;
        #pragma unroll
        for (int k = 0; k < 64; ++k) { float d = crow[k] - mval; vv = fmaf(d, d, vv); }
        vv *= (1.0f / 64.0f);
        s_mean[tid] = mval;
        s_rstd[tid] = rsqrtf(vv + LN_EPS);
    }
    __syncthreads();
    #pragma unroll
    for (int e = 0; e < 16; ++e) {
        int idx = e * THREADS + tid;
        int b = idx >> 6, k = idx & 63;
        int gb = row0 + b;
        if (gb < batch) {
            float cv = cmat[b * 65 + k];
            float y  = (cv - s_mean[b]) * s_rstd[b] * s_gam[k] + s_bet[k];
            float g  = 0.5f * y * (1.0f + erff(y * 0.70710678118654752f));
            out[(size_t)gb * WID + k] = g;
        }
    }
}

extern "C" void kernel_launch(void* const* d_in, const int* in_sizes, int n_in,
                              void* d_out, int out_size, void* d_ws, size_t ws_size,
                              hipStream_t stream) {
    (void)n_in; (void)out_size; (void)d_ws; (void)ws_size;
    const float* inputs   = (const float*)d_in[0];
    const float* shift    = (const float*)d_in[1];
    const float* logscale = (const float*)d_in[2];
    const float* coeffs   = (const float*)d_in[3];
    const float* skip_Wp  = (const float*)d_in[4];
    const float* skip_bp  = (const float*)d_in[5];
    const float* bias     = (const float*)d_in[6];
    const float* gamma_   = (const float*)d_in[7];
    const float* beta_    = (const float*)d_in[8];
    // d_in[9] = knots: recomputed analytically on device
    float* out = (float*)d_out;

    int batch = in_sizes[0] / IN_F;
    int grid  = (batch + ROWS_PER_WG - 1) / ROWS_PER_WG;
    kan_fused_kernel<<<grid, THREADS, SMEM_BYTES, stream>>>(
        inputs, shift, logscale, coeffs, skip_Wp, skip_bp, bias, gamma_, beta_,
        out, batch);
}